// Map3D_81020263072022
// MI455X (gfx1250) — compile-verified
//
#include <hip/hip_runtime.h>
#include <hip/hip_bf16.h>

typedef __attribute__((ext_vector_type(2))) float        v2f;
typedef __attribute__((ext_vector_type(8))) float        v8f;
typedef __attribute__((ext_vector_type(4))) unsigned int v4u;
typedef __attribute__((ext_vector_type(4))) int          v4i;
typedef __attribute__((ext_vector_type(8))) int          v8i;

#define SIZE   48
#define PAD    2
#define PADDED 52
#define PVOL   (PADDED*PADDED*PADDED)   /* 140608 */
#define IVOL   (SIZE*SIZE*SIZE)         /* 110592 */
#define N1     1024
#define N2     128
#define N3     10
#define NIN    2048
#define CONVEY 0.9f
#define KSPAN  (IVOL/16)                /* 6912 floats per K-chunk */

// workspace layout in floats
#define WS_BUFA 0
#define WS_BUFB (PVOL)
#define WS_X    (2*PVOL)
#define WS_Y1   (2*PVOL + IVOL)
#define WS_Y2   (2*PVOL + IVOL + N1)

// ---------------------------------------------------------------------------
// TDM: stage a contiguous run of `nfloats` f32 from global into LDS.
// D# per CDNA5 ISA: group0 = {count=1 | lds_addr |
// global_addr[56:0] | type=2}; group1 = {data_size=4B, tensor_dim0=n,
// tensor_dim1=1, tile_dim0=n, tile_dim1=1, dim0_stride=n}; groups 2/3 zero
// (2D tensor). Tracked by TENSORcnt.
// ---------------------------------------------------------------------------
__device__ __forceinline__ void tdm_copy_1d_to_lds(unsigned lds_addr,
                                                   const void* gptr,
                                                   unsigned nfloats) {
    unsigned long long ga = (unsigned long long)(uintptr_t)gptr;
    v4u g0;
    g0.x = 0x1u;                                   // count=1, user descriptor
    g0.y = lds_addr;                               // lds_addr [63:32]
    g0.z = (unsigned)ga;                           // global_addr [95:64]
    g0.w = (unsigned)((ga >> 32) & 0x01FFFFFFull)  // global_addr [120:96]
         | (2u << 30);                             // type=2 ("image") [127:126]
    v8i g1;
    g1[0] = (int)(2u << 16);                       // data_size=2 -> 4 bytes
    g1[1] = (int)((nfloats & 0xFFFFu) << 16);      // tensor_dim0[15:0] @ bits 63:48
    g1[2] = (int)((nfloats >> 16) | (1u << 16));   // tensor_dim0[31:16] | tensor_dim1=1
    g1[3] = (int)((nfloats & 0xFFFFu) << 16);      // tile_dim0 @ bits 127:112
    g1[4] = 1;                                     // tile_dim1=1
    g1[5] = (int)nfloats;                          // tensor_dim0_stride[31:0]
    g1[6] = 0;
    g1[7] = 0;
    v4i gz = {0, 0, 0, 0};
#if __has_include(<hip/amd_detail/amd_gfx1250_TDM.h>)
    v8i gz8 = {0, 0, 0, 0, 0, 0, 0, 0};
    __builtin_amdgcn_tensor_load_to_lds(g0, g1, gz, gz, gz8, 0);
#else
    __builtin_amdgcn_tensor_load_to_lds(g0, g1, gz, gz, 0);
#endif
}

// ---------------------------------------------------------------------------
// init: bufA <- neuron_matrix input, bufB <- 0, y1/y2 accumulators <- 0
// (every launch; ws is poisoned by the harness, atomics accumulate)
// ---------------------------------------------------------------------------
__global__ void nn3d_init_kernel(const float* __restrict__ nm_in,
                                 float* __restrict__ bufA,
                                 float* __restrict__ bufB,
                                 float* __restrict__ y1,
                                 float* __restrict__ y2) {
    int i = blockIdx.x * blockDim.x + threadIdx.x;
    if (i < PVOL) {
        bufA[i] = nm_in[i];
        bufB[i] = 0.0f;
    }
    if (i < N1) y1[i] = 0.0f;
    if (i < N2) y2[i] = 0.0f;
}

// ---------------------------------------------------------------------------
// scatter-add input injection (indices may collide -> atomics, matches .add)
// ---------------------------------------------------------------------------
__global__ void nn3d_scatter_kernel(const float* __restrict__ vinput,
                                    const int* __restrict__ node_idx,
                                    float* __restrict__ bufA) {
    int t = blockIdx.x * blockDim.x + threadIdx.x;
    if (t >= NIN) return;
    int x = node_idx[3*t + 0] + PAD;
    int y = node_idx[3*t + 1] + PAD;
    int z = node_idx[3*t + 2] + PAD;
    atomicAdd(&bufA[(x*PADDED + y)*PADDED + z], vinput[t]);
}

// ---------------------------------------------------------------------------
// one recurrent step: per-site unique-weight 4x4x4 neighborhood reduce.
// Bandwidth-bound on the synapse stream (256 B/site, L2-resident after step 1).
// Synapse is 256B-aligned per site -> float4 (global_load_b128) streaming.
// ---------------------------------------------------------------------------
__global__ void nn3d_step_kernel(const float* __restrict__ nmIn,
                                 const float* __restrict__ syn,
                                 float* __restrict__ nmOut) {
    int t = blockIdx.x * blockDim.x + threadIdx.x;
    if (t >= IVOL) return;
    int z = t % SIZE;
    int y = (t / SIZE) % SIZE;
    int x = t / (SIZE * SIZE);

    const float4* s4 = (const float4*)(syn + (size_t)t * 64);
    float acc = 0.0f;
#pragma unroll
    for (int a = 0; a < 4; ++a) {
        const float* plane = nmIn + ((x + a) * PADDED + y) * PADDED + z;
#pragma unroll
        for (int b = 0; b < 4; ++b) {
            const float* r = plane + b * PADDED;
            float4 w = s4[a * 4 + b];
            acc += w.x * r[0] + w.y * r[1] + w.z * r[2] + w.w * r[3];
        }
    }
    float v = acc > 0.0f ? acc * CONVEY : 0.0f;
    nmOut[((x + PAD) * PADDED + (y + PAD)) * PADDED + (z + PAD)] = v;
}

// ---------------------------------------------------------------------------
// flatten interior of final padded volume -> x[110592]
// ---------------------------------------------------------------------------
__global__ void nn3d_flatten_kernel(const float* __restrict__ nm,
                                    float* __restrict__ xflat) {
    int t = blockIdx.x * blockDim.x + threadIdx.x;
    if (t >= IVOL) return;
    int z = t % SIZE;
    int y = (t / SIZE) % SIZE;
    int x = t / (SIZE * SIZE);
    xflat[t] = nm[((x + PAD) * PADDED + (y + PAD)) * PADDED + (z + PAD)];
}

// ---------------------------------------------------------------------------
// Layer 1 GEMV via V_WMMA_F32_16X16X4_F32 with TDM-staged x in LDS.
// y1[n] = sum_k x[k] * W1[k*1024 + n]   (453 MB W1 stream = dominant cost)
// 1024 waves: wave = (ntile 0..63, kchunk 0..15); kchunk is block-uniform
// (8 waves/block), so each block TDM-loads its x K-chunk (27.6 KB) into LDS
// once; A-operands then come from LDS (ds_load). W1 loads use a bumped base
// pointer + immediate offsets (unroll 4 K-steps -> one v_add_u64 per 16 K).
// A 16x4 layout: VGPR0 holds K=2*half, VGPR1 K=2*half+1, M=lane%16 (row 0 only).
// C layout: VGPR0 lanes 0..15 = (M=0, N=lane) -> the GEMV result row.
// ---------------------------------------------------------------------------
__global__ void nn3d_gemv_w1_wmma(const float* __restrict__ x,
                                  const float* __restrict__ W1,
                                  float* __restrict__ y1) {
    __shared__ float xs[KSPAN];

    const int gwave = blockIdx.x * (blockDim.x >> 5) + (threadIdx.x >> 5);
    const int lane  = threadIdx.x & 31;
    const int half  = lane >> 4;   // 0 => lanes 0-15, 1 => lanes 16-31
    const int lm    = lane & 15;
    const int ntile = gwave & 63;  // 64 tiles of 16 columns
    const int kch   = gwave >> 6;  // 16 K-chunks; uniform within a block
    const int kbeg  = kch * KSPAN;
    const int n0    = ntile * 16;

    if (threadIdx.x == 0) {
        tdm_copy_1d_to_lds((unsigned)(uintptr_t)xs, x + kbeg, KSPAN);
        __builtin_amdgcn_s_wait_tensorcnt(0);
    }
    __syncthreads();

    // per-lane W1 stream pointer: row kbeg+2*half, column n0+lm
    const float* p = W1 + (size_t)(kbeg + 2 * half) * N1 + n0 + lm;
    int kloc = 2 * half;
    v8f acc = {};
    for (int it = 0; it < KSPAN / 16; ++it) {
#pragma unroll
        for (int u = 0; u < 4; ++u) {
            v2f a, b;
            float x0 = xs[kloc + 4 * u];          // ds_load, block-shared x
            float x1 = xs[kloc + 4 * u + 1];
            a.x = (lm == 0) ? x0 : 0.0f;          // A row M=0 only
            a.y = (lm == 0) ? x1 : 0.0f;
            b.x = p[4 * u * N1];                  // imm offsets 0/16K/32K/48K
            b.y = p[4 * u * N1 + N1];             // imm offsets +4K
            acc = __builtin_amdgcn_wmma_f32_16x16x4_f32(
                /*neg_a=*/false, a, /*neg_b=*/false, b,
                /*c_mod=*/(short)0, acc, /*reuse_a=*/false, /*reuse_b=*/false);
        }
        p += 16 * N1;                             // single pointer bump / 16 K
        kloc += 16;
        __builtin_prefetch(p + 32 * N1, 0, 1);    // stay ~128 KB ahead in L2
    }
    if (half == 0) atomicAdd(&y1[n0 + lm], acc[0]);  // M=0 row of C
}

// h1 = relu(y1 + b1), in place
__global__ void nn3d_bias_relu1(float* __restrict__ y1,
                                const float* __restrict__ b1) {
    int n = blockIdx.x * blockDim.x + threadIdx.x;
    if (n >= N1) return;
    float v = y1[n] + b1[n];
    y1[n] = v > 0.0f ? v : 0.0f;
}

// layer 2 partial sums: 8 blocks x 128 threads, K split across blocks,
// coalesced W2 rows, f32 atomic accumulation into y2
__global__ void nn3d_layer2_partial(const float* __restrict__ h1,
                                    const float* __restrict__ W2,
                                    float* __restrict__ y2) {
    int n  = threadIdx.x;          // 0..127
    int kb = blockIdx.x;           // 0..7
    float acc = 0.0f;
    int kend = kb * 128 + 128;
    for (int k = kb * 128; k < kend; ++k)
        acc += h1[k] * W2[k * N2 + n];
    atomicAdd(&y2[n], acc);
}

// h2 = relu(y2 + b2), in place
__global__ void nn3d_bias_relu2(float* __restrict__ y2,
                                const float* __restrict__ b2) {
    int n = threadIdx.x;
    if (n >= N2) return;
    float v = y2[n] + b2[n];
    y2[n] = v > 0.0f ? v : 0.0f;
}

// out = h2 @ W3 + b3
__global__ void nn3d_layer3(const float* __restrict__ h2,
                            const float* __restrict__ W3,
                            const float* __restrict__ b3,
                            float* __restrict__ out) {
    int j = threadIdx.x;
    if (j >= N3) return;
    float acc = 0.0f;
    for (int k = 0; k < N2; ++k)
        acc += h2[k] * W3[k * N3 + j];
    out[j] = acc + b3[j];
}

extern "C" void kernel_launch(void* const* d_in, const int* in_sizes, int n_in,
                              void* d_out, int out_size, void* d_ws, size_t ws_size,
                              hipStream_t stream) {
    const float* nm_in   = (const float*)d_in[0];  // neuron_matrix [52^3]
    const float* vinput  = (const float*)d_in[1];  // [2048]
    const float* synapse = (const float*)d_in[2];  // [48^3 * 64]
    const float* W1      = (const float*)d_in[3];  // [110592,1024]
    const float* b1      = (const float*)d_in[4];  // [1024]
    const float* W2      = (const float*)d_in[5];  // [1024,128]
    const float* b2      = (const float*)d_in[6];  // [128]
    const float* W3      = (const float*)d_in[7];  // [128,10]
    const float* b3      = (const float*)d_in[8];  // [10]
    const int*   nidx    = (const int*)d_in[9];    // [2048,3]
    float* out = (float*)d_out;

    float* ws   = (float*)d_ws;
    float* bufA = ws + WS_BUFA;
    float* bufB = ws + WS_BUFB;
    float* xfl  = ws + WS_X;
    float* y1   = ws + WS_Y1;
    float* y2   = ws + WS_Y2;

    // init ping-pong volumes + accumulators
    nn3d_init_kernel<<<(PVOL + 255) / 256, 256, 0, stream>>>(nm_in, bufA, bufB, y1, y2);
    // inject inputs
    nn3d_scatter_kernel<<<(NIN + 255) / 256, 256, 0, stream>>>(vinput, nidx, bufA);

    // 5 recurrent steps, ping-pong (final interior ends in `cur`)
    float* cur = bufA;
    float* nxt = bufB;
    for (int s = 0; s < 5; ++s) {
        nn3d_step_kernel<<<(IVOL + 255) / 256, 256, 0, stream>>>(cur, synapse, nxt);
        float* t = cur; cur = nxt; nxt = t;
    }
    nn3d_flatten_kernel<<<(IVOL + 255) / 256, 256, 0, stream>>>(cur, xfl);

    // layer 1: 1024 waves = 128 blocks x 256 threads (8 waves/block)
    nn3d_gemv_w1_wmma<<<128, 256, 0, stream>>>(xfl, W1, y1);
    nn3d_bias_relu1<<<(N1 + 255) / 256, 256, 0, stream>>>(y1, b1);
    nn3d_layer2_partial<<<8, N2, 0, stream>>>(y1, W2, y2);
    nn3d_bias_relu2<<<1, N2, 0, stream>>>(y2, b2);
    nn3d_layer3<<<1, 32, 0, stream>>>(y2, W3, b3, out);
}